// SS2D_region_28647431864582
// MI455X (gfx1250) — compile-verified
//
#include <hip/hip_runtime.h>
#include <math.h>

typedef float v2f __attribute__((ext_vector_type(2)));
typedef float v8f __attribute__((ext_vector_type(8)));

#define D_MODEL   96
#define D_INNER   192
#define D_STATE   16
#define DT_RANK   6
#define K_SPATIAL 4
#define B_SZ      4
#define H_IMG     64
#define W_IMG     64
#define L_SEQ     4096
#define XPROJ_C   (DT_RANK + 2 * D_STATE)   // 38
#define DC_INNER  4
#define KC        2
#define DC_STATE  16
#define DTC_RANK  6

// ---------------------------------------------------------------------------
// Async global->LDS 16-byte copy (ASYNCcnt-tracked) via the raw gfx1250
// instruction.  VDST = per-lane LDS byte offset (low 32 bits of the generic
// pointer to a __shared__ object ARE the LDS offset on AMDGPU), VADDR = 64-bit
// global address (GV mode, SADDR = off).
// ---------------------------------------------------------------------------
__device__ __forceinline__ void async_copy_b128(const float* gsrc, float* lds_dst) {
  unsigned ldsoff = (unsigned)(size_t)lds_dst;          // low 32 bits = LDS offset
  unsigned long long ga = (unsigned long long)(size_t)gsrc;
  asm volatile("global_load_async_to_lds_b128 %0, %1, off"
               :: "v"(ldsoff), "v"(ga) : "memory");
}

__device__ __forceinline__ void wait_async0() {
  asm volatile("s_wait_asynccnt 0x0" ::: "memory");
}

// ---------------------------------------------------------------------------
// Row-major WMMA GEMM:  C[M,N] = A[M,K] * W[N,K]^T   (f32, 16x16x4).
// 8 waves / 256 threads per block; block computes a 128x16 tile. The shared
// 16xK weight panel is staged into LDS once per block with async b128 copies
// (row-padded by 4 floats to spread LDS banks), then each wave reads its
// B-fragments from LDS (ds_load_b64) while streaming A from global.
// Fragment layouts per CDNA5 ISA 7.12.2:
//   A/B frag lane t: m|n = t&15, k-pair = (t>>4)*2  (VGPR0=K0/K2, VGPR1=K1/K3)
//   D   frag lane t: n = t&15, rows m = v + 8*(t>>4), v=0..7
// ---------------------------------------------------------------------------
__global__ __launch_bounds__(256) void gemm_rm_wmma(const float* __restrict__ A,
                                                    const float* __restrict__ W,
                                                    float* __restrict__ C,
                                                    int K, int N) {
  __shared__ __align__(16) float panel[16 * (192 + 4)];
  const int tid = threadIdx.x;
  const int wave = tid >> 5;
  const int t = tid & 31;
  const int half = t >> 4;
  const int lane16 = t & 15;
  const int n0 = blockIdx.y * 16;
  const int ldp = K + 4;                       // padded panel row stride
  // --- stage W panel (16 rows x K) into LDS asynchronously ---
  {
    const int cpr = K >> 2;                    // 16B chunks per row
    for (int i = tid; i < 16 * cpr; i += 256) {
      const int r = i / cpr, ci = i - r * cpr;
      async_copy_b128(W + (size_t)(n0 + r) * K + ci * 4,
                      &panel[r * ldp + ci * 4]);
    }
    wait_async0();
  }
  __syncthreads();
  const int mrow = blockIdx.x * 128 + wave * 16 + lane16;
  const float* Arow = A + (size_t)mrow * K;
  const float* Bpan = panel + lane16 * ldp;
  v8f acc = {};
  for (int kb = 0; kb < K; kb += 4) {
    if ((kb & 31) == 0)
      __builtin_prefetch(Arow + kb + 32, 0, 0);   // -> global_prefetch_b8
    const int ka = kb + half * 2;
    v2f a; a.x = Arow[ka]; a.y = Arow[ka + 1];
    v2f b; b.x = Bpan[ka]; b.y = Bpan[ka + 1];    // ds_load_b64
    acc = __builtin_amdgcn_wmma_f32_16x16x4_f32(false, a, false, b,
                                                (short)0, acc, false, false);
  }
  float* Crow = C + (size_t)(blockIdx.x * 128 + wave * 16 + half * 8) * N
                  + n0 + lane16;
#pragma unroll
  for (int v = 0; v < 8; ++v) Crow[(size_t)v * N] = acc[v];
}

// ---------------------------------------------------------------------------
// Depthwise 3x3 conv + bias + SiLU.  xz is (B,H,W,384) row-major; channels
// 0..191 are xh.  Output xc is channel-major (B, 192, H*W).
// ---------------------------------------------------------------------------
__global__ void dwconv_silu(const float* __restrict__ xz,
                            const float* __restrict__ w,
                            const float* __restrict__ bias,
                            float* __restrict__ xc) {
  int id = blockIdx.x * blockDim.x + threadIdx.x;
  if (id >= B_SZ * D_INNER * L_SEQ) return;
  const int wp = id % W_IMG;
  const int hp = (id / W_IMG) % H_IMG;
  const int c  = (id / L_SEQ) % D_INNER;
  const int b  = id / (L_SEQ * D_INNER);
  const float* wk = w + c * 9;
  float acc = bias[c];
#pragma unroll
  for (int dy = 0; dy < 3; ++dy) {
    const int hh = hp + dy - 1;
    if (hh < 0 || hh >= H_IMG) continue;
#pragma unroll
    for (int dx = 0; dx < 3; ++dx) {
      const int ww = wp + dx - 1;
      if (ww < 0 || ww >= W_IMG) continue;
      acc += wk[dy * 3 + dx] *
             xz[((size_t)(b * H_IMG + hh) * W_IMG + ww) * (2 * D_INNER) + c];
    }
  }
  xc[id] = acc / (1.0f + __expf(-acc));
}

// ---------------------------------------------------------------------------
// Stable two-way partition of pixels by mask: dest[p] = output slot of pixel p
// (background mask==0 first, original order preserved within class).
// One 1024-thread workgroup per batch; LDS Hillis-Steele scan over 4096.
// ---------------------------------------------------------------------------
__global__ __launch_bounds__(1024) void mask_rank(const int* __restrict__ xm,
                                                  int* __restrict__ dest) {
  __shared__ int bufA[1024];
  __shared__ int bufB[1024];
  const int b = blockIdx.x;
  const int t = threadIdx.x;
  const int* m = xm + (size_t)b * L_SEQ;
  const int base = t * 4;
  int mv[4];
  int local = 0;
#pragma unroll
  for (int i = 0; i < 4; ++i) { mv[i] = (m[base + i] != 0); local += mv[i]; }
  int* src = bufA;
  int* dst = bufB;
  src[t] = local;
  __syncthreads();
  for (int off = 1; off < 1024; off <<= 1) {
    int v = src[t];
    if (t >= off) v += src[t - off];
    dst[t] = v;
    __syncthreads();
    int* tmp = src; src = dst; dst = tmp;
  }
  const int incl = src[t];
  const int total_ones = src[1023];
  int ones_before = incl - local;   // exclusive prefix at element `base`
  const int zeros_total = L_SEQ - total_ones;
#pragma unroll
  for (int i = 0; i < 4; ++i) {
    const int p = base + i;
    dest[(size_t)b * L_SEQ + p] =
        mv[i] ? (zeros_total + ones_before) : (p - ones_before);
    ones_before += mv[i];
  }
}

// ---------------------------------------------------------------------------
// Build 4-direction stack xs (B,4,192,L):
//  k=0 region-sorted, k=1 HW-transposed, k=2 flip(k=0), k=3 flip(k=1).
// ---------------------------------------------------------------------------
__global__ void build_xs(const float* __restrict__ xc,
                         const int* __restrict__ dest,
                         float* __restrict__ xs) {
  int id = blockIdx.x * blockDim.x + threadIdx.x;
  if (id >= B_SZ * D_INNER * L_SEQ) return;
  const int p = id % L_SEQ;
  const int d = (id / L_SEQ) % D_INNER;
  const int b = id / (L_SEQ * D_INNER);
  const float v = xc[id];
  const int j  = dest[(size_t)b * L_SEQ + p];
  const int hh = p / W_IMG, ww = p % W_IMG;
  const int lt = ww * H_IMG + hh;
  float* base = xs + (size_t)b * 4 * D_INNER * L_SEQ;
  base[((size_t)(0 * D_INNER + d)) * L_SEQ + j]                 = v;
  base[((size_t)(1 * D_INNER + d)) * L_SEQ + lt]                = v;
  base[((size_t)(2 * D_INNER + d)) * L_SEQ + (L_SEQ - 1 - j)]   = v;
  base[((size_t)(3 * D_INNER + d)) * L_SEQ + (L_SEQ - 1 - lt)]  = v;
}

// ---------------------------------------------------------------------------
// x_dbl = einsum('bkdl,kcd->bkcl') as WMMA GEMM with column-major A and C:
//   A(m=l, k=d) = xs[bk, d, l]   (stride L along k)
//   W = x_proj_weight[kdir]  (38 x 192 row-major), N padded to 48 (guarded)
// grid = (L/16, 3, 16), block = 32.
// ---------------------------------------------------------------------------
__global__ __launch_bounds__(32) void gemm_xdbl_wmma(const float* __restrict__ xs,
                                                     const float* __restrict__ xpw,
                                                     float* __restrict__ xdbl) {
  const int m0 = blockIdx.x * 16;
  const int n0 = blockIdx.y * 16;
  const int bk = blockIdx.z;
  const int kdir = bk & 3;
  const int t = threadIdx.x;
  const int half = t >> 4;
  const int lane16 = t & 15;
  const float* A  = xs + (size_t)bk * D_INNER * L_SEQ;
  const float* Wm = xpw + (size_t)kdir * XPROJ_C * D_INNER;
  const int n = n0 + lane16;
  v8f acc = {};
  for (int kb = 0; kb < D_INNER; kb += 4) {
    const int kk = kb + half * 2;
    v2f a;
    a.x = A[(size_t)kk * L_SEQ + m0 + lane16];
    a.y = A[(size_t)(kk + 1) * L_SEQ + m0 + lane16];
    v2f bfr;
    float w0 = 0.f, w1 = 0.f;
    if (n < XPROJ_C) {
      const float* wp = Wm + (size_t)n * D_INNER + kk;
      w0 = wp[0]; w1 = wp[1];
    }
    bfr.x = w0; bfr.y = w1;
    acc = __builtin_amdgcn_wmma_f32_16x16x4_f32(false, a, false, bfr,
                                                (short)0, acc, false, false);
  }
  if (n < XPROJ_C) {
    float* Obase = xdbl + (size_t)bk * XPROJ_C * L_SEQ
                        + (size_t)n * L_SEQ + m0 + half * 8;
#pragma unroll
    for (int v = 0; v < 8; ++v) Obase[v] = acc[v];
  }
}

// ---------------------------------------------------------------------------
// Selective scan (spatial): 16 workgroups (one per (b,kdir)), 192 lanes (one
// per d), 16 states in VGPRs. dt low-rank -> full projection folded in.
// Per step: 38 shared values staged via LDS, ~16 v_exp_f32 per lane.
// ---------------------------------------------------------------------------
__global__ __launch_bounds__(192) void selective_scan_spatial(
    const float* __restrict__ xs, const float* __restrict__ xdbl,
    const float* __restrict__ dtw, const float* __restrict__ dtb,
    const float* __restrict__ Alogs, const float* __restrict__ Ds,
    float* __restrict__ outy) {
  __shared__ float srow[XPROJ_C];
  const int bk = blockIdx.x;       // b*4 + kdir
  const int kdir = bk & 3;
  const int d = threadIdx.x;
  const float* u  = xs  + ((size_t)bk * D_INNER + d) * L_SEQ;
  const float* xd = xdbl + (size_t)bk * XPROJ_C * L_SEQ;
  float* yo = outy + ((size_t)bk * D_INNER + d) * L_SEQ;
  const int gd = kdir * D_INNER + d;
  float A[D_STATE];
#pragma unroll
  for (int n = 0; n < D_STATE; ++n) A[n] = -__expf(Alogs[gd * D_STATE + n]);
  float wr[DT_RANK];
#pragma unroll
  for (int r = 0; r < DT_RANK; ++r) wr[r] = dtw[gd * DT_RANK + r];
  const float bias = dtb[gd];
  const float Dd = Ds[gd];
  float h[D_STATE];
#pragma unroll
  for (int n = 0; n < D_STATE; ++n) h[n] = 0.f;
  for (int l = 0; l < L_SEQ; ++l) {
    if (d < XPROJ_C) srow[d] = xd[(size_t)d * L_SEQ + l];
    __syncthreads();
    float dt = bias;
#pragma unroll
    for (int r = 0; r < DT_RANK; ++r) dt += wr[r] * srow[r];
    const float delta = (dt > 20.f) ? dt : __logf(1.f + __expf(dt));
    const float ul = u[l];
    const float du = delta * ul;
    float y = 0.f;
#pragma unroll
    for (int n = 0; n < D_STATE; ++n) {
      h[n] = h[n] * __expf(delta * A[n]) + du * srow[DT_RANK + n];
      y += h[n] * srow[DT_RANK + D_STATE + n];
    }
    yo[l] = y + ul * Dd;
    __syncthreads();
  }
}

// ---------------------------------------------------------------------------
// Merge 4 directions: y = y0 + flip(y2) + transHW(y1) + transHW(flip(y3)),
// stored (B, L, 192) row-major for LN + final GEMM.
// ---------------------------------------------------------------------------
__global__ void merge_y(const float* __restrict__ outy, float* __restrict__ ym) {
  int id = blockIdx.x * blockDim.x + threadIdx.x;
  if (id >= B_SZ * D_INNER * L_SEQ) return;
  const int l = id % L_SEQ;
  const int d = (id / L_SEQ) % D_INNER;
  const int b = id / (L_SEQ * D_INNER);
  const float* base = outy + (size_t)b * 4 * D_INNER * L_SEQ;
  const int hh = l / W_IMG, ww = l % W_IMG;
  const int src = ww * H_IMG + hh;
  const float v =
      base[((size_t)(0 * D_INNER + d)) * L_SEQ + l] +
      base[((size_t)(2 * D_INNER + d)) * L_SEQ + (L_SEQ - 1 - l)] +
      base[((size_t)(1 * D_INNER + d)) * L_SEQ + src] +
      base[((size_t)(3 * D_INNER + d)) * L_SEQ + (L_SEQ - 1 - src)];
  ym[((size_t)b * L_SEQ + l) * D_INNER + d] = v;
}

// ---------------------------------------------------------------------------
// Channel branch: pool -> tiny 2-direction scan over L=192 -> catt (B,192).
// ---------------------------------------------------------------------------
__global__ __launch_bounds__(192) void channel_branch(
    const float* __restrict__ xc, const float* __restrict__ cinw,
    const float* __restrict__ cinb, const float* __restrict__ xcpw,
    const float* __restrict__ dtcw, const float* __restrict__ dtcb,
    const float* __restrict__ Aclogs, const float* __restrict__ Dsc,
    const float* __restrict__ coutw, const float* __restrict__ coutb,
    float* __restrict__ catt) {
  __shared__ float xs_c[KC][DC_INNER][D_INNER];
  __shared__ float yc[KC][DC_INNER][D_INNER];
  const int b = blockIdx.x;
  const int t = threadIdx.x;
  const float* xcb = xc + ((size_t)b * D_INNER + t) * L_SEQ;
  float sum = 0.f;
  for (int l = 0; l < L_SEQ; ++l) sum += xcb[l];
  const float pooled = sum * (1.0f / (float)L_SEQ);
#pragma unroll
  for (int i = 0; i < DC_INNER; ++i) {
    const float cv = pooled * cinw[i] + cinb[i];
    xs_c[0][i][t] = cv;
    xs_c[1][i][D_INNER - 1 - t] = cv;
  }
  __syncthreads();
  if (t < KC * DC_INNER) {
    const int k = t / DC_INNER, d = t % DC_INNER;
    const int gd = k * DC_INNER + d;
    const float* Wk = xcpw + (size_t)k * (DTC_RANK + 2 * DC_STATE) * DC_INNER;
    float A[DC_STATE];
#pragma unroll
    for (int n = 0; n < DC_STATE; ++n) A[n] = -__expf(Aclogs[gd * DC_STATE + n]);
    float wr[DTC_RANK];
#pragma unroll
    for (int r = 0; r < DTC_RANK; ++r) wr[r] = dtcw[gd * DTC_RANK + r];
    const float bias = dtcb[gd];
    const float Dd = Dsc[gd];
    float h[DC_STATE];
#pragma unroll
    for (int n = 0; n < DC_STATE; ++n) h[n] = 0.f;
    for (int l = 0; l < D_INNER; ++l) {
      float xi[DC_INNER];
#pragma unroll
      for (int i = 0; i < DC_INNER; ++i) xi[i] = xs_c[k][i][l];
      float dt = bias;
#pragma unroll
      for (int r = 0; r < DTC_RANK; ++r) {
        float pr = 0.f;
#pragma unroll
        for (int i = 0; i < DC_INNER; ++i) pr += xi[i] * Wk[r * DC_INNER + i];
        dt += wr[r] * pr;
      }
      const float delta = (dt > 20.f) ? dt : __logf(1.f + __expf(dt));
      const float ul = xi[d];
      const float du = delta * ul;
      float y = 0.f;
#pragma unroll
      for (int n = 0; n < DC_STATE; ++n) {
        float Bn = 0.f, Cn = 0.f;
#pragma unroll
        for (int i = 0; i < DC_INNER; ++i) {
          Bn += xi[i] * Wk[(DTC_RANK + n) * DC_INNER + i];
          Cn += xi[i] * Wk[(DTC_RANK + DC_STATE + n) * DC_INNER + i];
        }
        h[n] = h[n] * __expf(delta * A[n]) + du * Bn;
        y += h[n] * Cn;
      }
      yc[k][d][l] = y + ul * Dd;
    }
  }
  __syncthreads();
  float acc = coutb[0];
#pragma unroll
  for (int i = 0; i < DC_INNER; ++i)
    acc += (yc[0][i][t] + yc[1][i][D_INNER - 1 - t]) * coutw[i];
  catt[(size_t)b * D_INNER + t] = acc;
}

// ---------------------------------------------------------------------------
// LN(y) over d  +  channel-attention apply + channel-norm over d  + SiLU gate:
//   yf = (LN(ym) + CN(xc*catt)) * silu(z).   One block per (b,l), 192 lanes.
// ---------------------------------------------------------------------------
__global__ __launch_bounds__(192) void combine_norm(
    const float* __restrict__ ym, const float* __restrict__ xc,
    const float* __restrict__ catt, const float* __restrict__ xz,
    const float* __restrict__ onw, const float* __restrict__ onb,
    const float* __restrict__ cnw, const float* __restrict__ cnb,
    float* __restrict__ yf) {
  __shared__ float rA[256];
  __shared__ float rB[256];
  const int bl = blockIdx.x;
  const int b = bl / L_SEQ, l = bl % L_SEQ;
  const int c = threadIdx.x;
  const float yv = ym[(size_t)bl * D_INNER + c];
  const float xch = xc[((size_t)b * D_INNER + c) * L_SEQ + l]
                    * catt[(size_t)b * D_INNER + c];
  rA[c] = yv; rB[c] = xch;
  if (c < 64) { rA[192 + c] = 0.f; rB[192 + c] = 0.f; }
  __syncthreads();
  for (int off = 128; off >= 1; off >>= 1) {
    if (c < off) { rA[c] += rA[c + off]; rB[c] += rB[c + off]; }
    __syncthreads();
  }
  const float muY = rA[0] * (1.0f / D_INNER);
  const float muX = rB[0] * (1.0f / D_INNER);
  __syncthreads();
  const float dy = yv - muY, dx = xch - muX;
  rA[c] = dy * dy; rB[c] = dx * dx;
  if (c < 64) { rA[192 + c] = 0.f; rB[192 + c] = 0.f; }
  __syncthreads();
  for (int off = 128; off >= 1; off >>= 1) {
    if (c < off) { rA[c] += rA[c + off]; rB[c] += rB[c + off]; }
    __syncthreads();
  }
  const float vY = rA[0] * (1.0f / D_INNER);
  const float vX = rB[0] * (1.0f / D_INNER);
  const float yln = dy * rsqrtf(vY + 1e-5f) * onw[c] + onb[c];
  const float xn  = dx * rsqrtf(vX + 1e-5f) * cnw[c] + cnb[c];
  const float zv = xz[(size_t)bl * (2 * D_INNER) + D_INNER + c];
  const float sz = zv / (1.0f + __expf(-zv));
  yf[(size_t)bl * D_INNER + c] = (yln + xn) * sz;
}

// ---------------------------------------------------------------------------
extern "C" void kernel_launch(void* const* d_in, const int* in_sizes, int n_in,
                              void* d_out, int out_size, void* d_ws, size_t ws_size,
                              hipStream_t stream) {
  (void)in_sizes; (void)n_in; (void)out_size; (void)ws_size;
  const float* x      = (const float*)d_in[0];
  const int*   xm     = (const int*)  d_in[1];
  const float* ipw    = (const float*)d_in[2];
  const float* convw  = (const float*)d_in[3];
  const float* convb  = (const float*)d_in[4];
  const float* xpw    = (const float*)d_in[5];
  const float* dtw    = (const float*)d_in[6];
  const float* dtb    = (const float*)d_in[7];
  const float* Alogs  = (const float*)d_in[8];
  const float* Dsv    = (const float*)d_in[9];
  const float* onw    = (const float*)d_in[10];
  const float* onb    = (const float*)d_in[11];
  const float* opw    = (const float*)d_in[12];
  const float* cinw   = (const float*)d_in[13];
  const float* cinb   = (const float*)d_in[14];
  const float* coutw  = (const float*)d_in[15];
  const float* coutb  = (const float*)d_in[16];
  const float* xcpw   = (const float*)d_in[17];
  const float* Dscv   = (const float*)d_in[18];
  const float* Aclogs = (const float*)d_in[19];
  const float* dtcw   = (const float*)d_in[20];
  const float* dtcb   = (const float*)d_in[21];
  const float* cnw    = (const float*)d_in[22];
  const float* cnb    = (const float*)d_in[23];
  float* out = (float*)d_out;

  const size_t M = (size_t)B_SZ * L_SEQ;          // 16384
  float* ws = (float*)d_ws;
  size_t off = 0;
  float* xz   = ws + off; off += M * (2 * D_INNER);                    // 6.3M
  float* xc   = ws + off; off += (size_t)B_SZ * D_INNER * L_SEQ;       // 3.1M
  int*   dest = (int*)(ws + off); off += (size_t)B_SZ * L_SEQ;         // 16K
  float* xs   = ws + off; off += (size_t)16 * D_INNER * L_SEQ;         // 12.6M
  float* xdbl = ws + off; off += (size_t)16 * XPROJ_C * L_SEQ;         // 2.5M
  float* outy = ws + off; off += (size_t)16 * D_INNER * L_SEQ;         // 12.6M
  float* ymb  = ws + off; off += M * D_INNER;                          // 3.1M
  float* catt = ws + off; off += (size_t)B_SZ * D_INNER;
  float* yf   = ws + off; off += M * D_INNER;                          // 3.1M

  const int elems = B_SZ * D_INNER * L_SEQ;

  // 1) in_proj GEMM: (16384 x 96) @ (384 x 96)^T -> xz
  gemm_rm_wmma<<<dim3(M / 128, (2 * D_INNER) / 16), 256, 0, stream>>>(
      x, ipw, xz, D_MODEL, 2 * D_INNER);
  // 2) depthwise conv + SiLU -> xc (B,192,L)
  dwconv_silu<<<(elems + 255) / 256, 256, 0, stream>>>(xz, convw, convb, xc);
  // 3) mask partition ranks -> dest
  mask_rank<<<B_SZ, 1024, 0, stream>>>(xm, dest);
  // 4) 4-direction stack -> xs
  build_xs<<<(elems + 255) / 256, 256, 0, stream>>>(xc, dest, xs);
  // 5) per-direction low-rank projection -> xdbl (WMMA, col-major A/C)
  gemm_xdbl_wmma<<<dim3(L_SEQ / 16, 3, 16), 32, 0, stream>>>(xs, xpw, xdbl);
  // 6) selective scan over L=4096
  selective_scan_spatial<<<16, 192, 0, stream>>>(xs, xdbl, dtw, dtb, Alogs,
                                                 Dsv, outy);
  // 7) merge directions -> ym (B,L,192)
  merge_y<<<(elems + 255) / 256, 256, 0, stream>>>(outy, ymb);
  // 8) channel branch -> catt (B,192)
  channel_branch<<<B_SZ, 192, 0, stream>>>(xc, cinw, cinb, xcpw, dtcw, dtcb,
                                           Aclogs, Dscv, coutw, coutb, catt);
  // 9) LN + channel-norm + gate -> yf
  combine_norm<<<M, 192, 0, stream>>>(ymb, xc, catt, xz, onw, onb, cnw, cnb, yf);
  // 10) out_proj GEMM: (16384 x 192) @ (96 x 192)^T -> out
  gemm_rm_wmma<<<dim3(M / 128, D_MODEL / 16), 256, 0, stream>>>(
      yf, opw, out, D_INNER, D_MODEL);
}